// ComposedFeatureTransformer_11682311045695
// MI455X (gfx1250) — compile-verified
//
#include <hip/hip_runtime.h>
#include <stdint.h>

#define L1        2048
#define NUM_OUT   2056
#define BATCH     8192
#define KACT      32
#define DEPTH     4            // async pipeline depth (LDS ring slots)
#define TPB       256
#define ROWF4     (L1 / 4)     // 512 float4 staged per weight row

typedef __attribute__((ext_vector_type(4))) float v4f;

__device__ __forceinline__ void fma4(float4& a, float s, float4 w) {
    a.x += s * w.x; a.y += s * w.y; a.z += s * w.z; a.w += s * w.w;
}

// Non-temporal 16B store (clang builtin requires native vector types).
__device__ __forceinline__ void nt_store4(float4 v, float4* p) {
    v4f x;
    x.x = v.x; x.y = v.y; x.z = v.z; x.w = v.w;
    __builtin_nontemporal_store(x, (v4f*)p);
}

// Stage one weight row (first 2048 floats) into an LDS slot.
// Each lane copies the two 16B chunks it will later consume, so no
// cross-thread dependency exists and no workgroup barrier is needed.
// s_wait_dscnt 0 first: guarantees our ds_loads of the slot being reused
// have completed before the async writes can land in it.
__device__ __forceinline__ void stage_row(uint32_t lds0, uint32_t lds1,
                                          uint32_t g0, uint32_t g1,
                                          const float* rowbase) {
    asm volatile(
        "s_wait_dscnt 0x0\n\t"
        "global_load_async_to_lds_b128 %0, %2, %4\n\t"
        "global_load_async_to_lds_b128 %1, %3, %4"
        :: "v"(lds0), "v"(lds1), "v"(g0), "v"(g1), "s"(rowbase)
        : "memory");
}

__global__ __launch_bounds__(TPB) void nnue_ft_kernel(
    const int*   __restrict__ idx0, const float* __restrict__ val0,
    const int*   __restrict__ idx1, const float* __restrict__ val1,
    const float* __restrict__ weight,
    const float* __restrict__ bias_ft,
    const float* __restrict__ bias_psqt,
    float*       __restrict__ out) {

    __shared__ float4 sh[DEPTH][ROWF4];   // 4 x 8 KB ring buffer

    const int tid = threadIdx.x;
    const int b   = blockIdx.x;
    const int p   = blockIdx.y;

    const int*   __restrict__ idx = p ? idx1 : idx0;   // uniform select
    const float* __restrict__ val = p ? val1 : val0;

    // LDS byte addresses (flat-address low dword == LDS offset on gfx1250)
    const uint32_t ldsbase = (uint32_t)(uintptr_t)(&sh[0][0]);
    const uint32_t l_t0 = ldsbase + (uint32_t)tid * 16u;
    const uint32_t l_t1 = l_t0 + (uint32_t)TPB * 16u;
    const uint32_t g0   = (uint32_t)tid * 16u;          // byte offset in row
    const uint32_t g1   = g0 + (uint32_t)TPB * 16u;

    // Accumulators initialized with bias (cols 4t, 1024+4t, psqt for lanes 0-1)
    float4 acc0 = ((const float4*)bias_ft)[tid];
    float4 acc1 = ((const float4*)bias_ft)[TPB + tid];
    float4 accp = make_float4(0.f, 0.f, 0.f, 0.f);
    if (tid < 2) accp = ((const float4*)bias_psqt)[tid];

    // ---- prologue: stage rows 0..DEPTH-1 (2 async b128 per stage per lane)
#pragma unroll
    for (int k = 0; k < DEPTH; ++k) {
        const float* row = weight + (size_t)idx[b * KACT + k] * NUM_OUT;
        const uint32_t soff = (uint32_t)k * (ROWF4 * 16u);
        stage_row(l_t0 + soff, l_t1 + soff, g0, g1, row);
    }

    // ---- main pipeline: consume row k, stage row k+DEPTH
#pragma unroll 4
    for (int k = 0; k < KACT - DEPTH; ++k) {
        // allow 2*(DEPTH-1)=6 outstanding -> oldest stage (row k) is complete
        asm volatile("s_wait_asynccnt 0x6" ::: "memory");
        const int slot = k & (DEPTH - 1);

        float4 w0 = sh[slot][tid];
        float4 w1 = sh[slot][TPB + tid];
        const float v = val[b * KACT + k];              // uniform (SGPR)
        fma4(acc0, v, w0);
        fma4(acc1, v, w1);
        if (tid < 2) {                                  // 8 psqt cols, direct
            const float* rowk = weight + (size_t)idx[b * KACT + k] * NUM_OUT;
            float4 wp = *(const float4*)(rowk + L1 + 4 * tid);
            fma4(accp, v, wp);
        }

        const int kn = k + DEPTH;                       // reuse the slot
        const float* rown = weight + (size_t)idx[b * KACT + kn] * NUM_OUT;
        const uint32_t soff = (uint32_t)slot * (ROWF4 * 16u);
        stage_row(l_t0 + soff, l_t1 + soff, g0, g1, rown);
    }

    // ---- drain: everything staged; wait for all async copies, consume tail
    asm volatile("s_wait_asynccnt 0x0" ::: "memory");
#pragma unroll
    for (int k = KACT - DEPTH; k < KACT; ++k) {
        const int slot = k & (DEPTH - 1);
        float4 w0 = sh[slot][tid];
        float4 w1 = sh[slot][TPB + tid];
        const float v = val[b * KACT + k];
        fma4(acc0, v, w0);
        fma4(acc1, v, w1);
        if (tid < 2) {
            const float* rowk = weight + (size_t)idx[b * KACT + k] * NUM_OUT;
            float4 wp = *(const float4*)(rowk + L1 + 4 * tid);
            fma4(accp, v, wp);
        }
    }

    // ---- write output row (fully covers all 2056 cols)
    // Non-temporal stores: output is write-once/stream, keep L2 capacity
    // for the weight gather table (~11.6x average row reuse).
    float4* orow = (float4*)(out + ((size_t)p * BATCH + (size_t)b) * NUM_OUT);
    nt_store4(acc0, &orow[tid]);
    nt_store4(acc1, &orow[TPB + tid]);
    if (tid < 2) nt_store4(accp, &orow[L1 / 4 + tid]);
}

extern "C" void kernel_launch(void* const* d_in, const int* in_sizes, int n_in,
                              void* d_out, int out_size, void* d_ws, size_t ws_size,
                              hipStream_t stream) {
    const int*   idx0      = (const int*)  d_in[0];
    const float* val0      = (const float*)d_in[1];
    const int*   idx1      = (const int*)  d_in[2];
    const float* val1      = (const float*)d_in[3];
    const float* weight    = (const float*)d_in[4];
    const float* bias_ft   = (const float*)d_in[5];
    const float* bias_psqt = (const float*)d_in[6];
    float*       out       = (float*)d_out;

    dim3 grid(BATCH, 2);
    nnue_ft_kernel<<<grid, dim3(TPB), 0, stream>>>(
        idx0, val0, idx1, val1, weight, bias_ft, bias_psqt, out);
}